// GraphProject_19799799234740
// MI455X (gfx1250) — compile-verified
//
#include <hip/hip_runtime.h>
#include <cstdint>

typedef float v2f __attribute__((ext_vector_type(2)));
typedef float v8f __attribute__((ext_vector_type(8)));

#define NVERT  100000
#define NBATCH 8
#define VPB    1024
#define NCHUNK ((NVERT + VPB - 1) / VPB)   // 98

#define OFF_F0 3
#define OFF_F1 67
#define OFF_F2 195
#define OFF_F3 451
#define OUTW   963

// transposed feature maps in workspace (float offsets), layout [B][H][W][C]
#define WS0 0
#define WS0_SZ (8 * 3136 * 64)
#define WS1 (WS0 + WS0_SZ)
#define WS1_SZ (8 * 784 * 128)
#define WS2 (WS1 + WS1_SZ)
#define WS2_SZ (8 * 196 * 256)
#define WS3 (WS2 + WS2_SZ)
#define WS3_SZ (8 * 49 * 512)

#define F3_PITCH 516                 // 512 + 4 pad words per pixel row (bank-conflict-free)
#define LDS_F3_WORDS (49 * F3_PITCH) // 25284 words = 101136 bytes

// ---------------- transpose [B][C][P] -> [B][P][C] ----------------
__global__ void transpose_chw_hwc(const float* __restrict__ in,
                                  float* __restrict__ out, int C, int P) {
  int total = NBATCH * C * P;
  for (int idx = blockIdx.x * blockDim.x + threadIdx.x; idx < total;
       idx += gridDim.x * blockDim.x) {
    int c = idx % C;          // fastest -> coalesced writes
    int rest = idx / C;
    int p = rest % P;
    int b = rest / P;
    out[idx] = in[(b * C + c) * P + p];
  }
}

// ---------------- bilinear over HWC-transposed map ----------------
__device__ __forceinline__ void bilin_scale(float* __restrict__ op,
                                            const float* __restrict__ srcT,
                                            int s, int C, float h, float w_,
                                            int lane) {
  float inv = (float)s / 224.0f;     // exact: 0.25, 0.125, 0.0625
  float xc = fminf(fmaxf(h * inv, 0.0f), (float)(s - 1));
  float yc = fminf(fmaxf(w_ * inv, 0.0f), (float)(s - 1));
  float x1f = floorf(xc), x2f = ceilf(xc);
  float y1f = floorf(yc), y2f = ceilf(yc);
  int x1 = (int)x1f, x2 = (int)x2f, y1 = (int)y1f, y2 = (int)y2f;
  float w11 = (x2f - xc) * (y2f - yc);
  float w12 = (x2f - xc) * (yc - y1f);
  float w21 = (xc - x1f) * (y2f - yc);
  float w22 = (xc - x1f) * (yc - y1f);
  const float* p11 = srcT + (x1 * s + y1) * C;
  const float* p12 = srcT + (x1 * s + y2) * C;
  const float* p21 = srcT + (x2 * s + y1) * C;
  const float* p22 = srcT + (x2 * s + y2) * C;
  for (int c0 = 0; c0 < C; c0 += 32) {
    int c = c0 + lane;
    op[c] = w11 * p11[c] + w21 * p21[c] + w12 * p12[c] + w22 * p22[c];
  }
}

// ---------------- main kernel ----------------
__global__ __launch_bounds__(256) void graphproject_kernel(
    const float* __restrict__ vertices, const float* __restrict__ ws,
    float* __restrict__ out) {
  __shared__ float lds_f3[LDS_F3_WORDS];

  const int b    = blockIdx.y;
  const int v0   = blockIdx.x * VPB;
  const int tid  = threadIdx.x;
  const int lane = tid & 31;
  const int wave = tid >> 5;

  // ---- Phase A: async-stage transposed feat3[b] (49 x 512) into LDS (pitch 516)
  {
    const float* src = ws + WS3 + (size_t)b * (49 * 512);
    for (int j = tid; j < 49 * 128; j += 256) {     // 128 x float4 per pixel row
      int p  = j >> 7;
      int c4 = (j & 127) << 2;
      uint32_t ldsaddr = (uint32_t)(uintptr_t)(&lds_f3[p * F3_PITCH + c4]);
      uint64_t gaddr   = (uint64_t)(uintptr_t)(src + p * 512 + c4);
      asm volatile("global_load_async_to_lds_b128 %0, %1, off"
                   :: "v"(ldsaddr), "v"(gaddr) : "memory");
    }
  }

  // ---- Phase B: xyz + feat0/1/2 (one wave per vertex, lanes over channels)
  const size_t outBatch = (size_t)b * NVERT;
  for (int i = wave; i < VPB; i += 8) {
    int v = v0 + i;
    if (v >= NVERT) break;
    const float* vp = vertices + (size_t)(b * NVERT + v) * 3;
    float X = vp[0], Y = vp[1], Z = vp[2];
    float h  = 250.0f * (Y / Z) + 112.0f;
    float w_ = 250.0f * (X / (-Z)) + 112.0f;
    float* op = out + (outBatch + v) * OUTW;
    if (lane < 3) op[lane] = vp[lane];
    bilin_scale(op + OFF_F0, ws + WS0 + (size_t)b * 3136 * 64, 56, 64, h, w_, lane);
    bilin_scale(op + OFF_F1, ws + WS1 + (size_t)b * 784 * 128, 28, 128, h, w_, lane);
    bilin_scale(op + OFF_F2, ws + WS2 + (size_t)b * 196 * 256, 14, 256, h, w_, lane);
  }

  asm volatile("s_wait_asynccnt 0" ::: "memory");
  __syncthreads();

  // ---- Phase C: feat3 as GEMM  D(16v x 16c) = W(16 x 52) x F(52 x 16) via
  //      13 chained v_wmma_f32_16x16x4_f32, B from LDS, A rebuilt per 16-vertex tile.
  for (int t = wave; t < VPB / 16; t += 8) {
    int vbase = v0 + t * 16;
    if (vbase >= NVERT) break;

    int m     = lane & 15;           // vertex row (A/M) and channel col (B/N)
    int half2 = (lane >> 4) << 1;    // K offset 0 or 2 within a K=4 step
    int v = vbase + m;
    const float* vp = vertices + (size_t)(b * NVERT + v) * 3;
    float X = vp[0], Y = vp[1], Z = vp[2];
    float h  = 250.0f * (Y / Z) + 112.0f;
    float w_ = 250.0f * (X / (-Z)) + 112.0f;
    float xc = fminf(fmaxf(h * 0.03125f, 0.0f), 6.0f);
    float yc = fminf(fmaxf(w_ * 0.03125f, 0.0f), 6.0f);
    float x1f = floorf(xc), x2f = ceilf(xc), y1f = floorf(yc), y2f = ceilf(yc);
    float w11 = (x2f - xc) * (y2f - yc);
    float w12 = (x2f - xc) * (yc - y1f);
    float w21 = (xc - x1f) * (y2f - yc);
    float w22 = (xc - x1f) * (yc - y1f);
    int i11 = (int)x1f * 7 + (int)y1f;
    int i12 = (int)x1f * 7 + (int)y2f;
    int i21 = (int)x2f * 7 + (int)y1f;
    int i22 = (int)x2f * 7 + (int)y2f;

    v2f A[13];
    int bofs0[13], bofs1[13];
#pragma unroll
    for (int ks = 0; ks < 13; ++ks) {
      int k0 = 4 * ks + half2;
      int k1 = k0 + 1;
      A[ks].x = (k0 == i11) ? w11 : (k0 == i12) ? w12 : (k0 == i21) ? w21
               : (k0 == i22) ? w22 : 0.0f;
      A[ks].y = (k1 == i11) ? w11 : (k1 == i12) ? w12 : (k1 == i21) ? w21
               : (k1 == i22) ? w22 : 0.0f;
      bofs0[ks] = (k0 > 48 ? 48 : k0) * F3_PITCH + m;  // pad K reads clamp; A==0 there
      bofs1[ks] = (k1 > 48 ? 48 : k1) * F3_PITCH + m;
    }

    int row = (lane >> 4) << 3;      // D: lanes16-31 hold rows M+8
#pragma unroll 1
    for (int ct = 0; ct < 32; ++ct) {
      int c0 = ct * 16;
      v8f acc = {};
#pragma unroll
      for (int ks = 0; ks < 13; ++ks) {
        v2f bm;
        bm.x = lds_f3[bofs0[ks] + c0];
        bm.y = lds_f3[bofs1[ks] + c0];
        acc = __builtin_amdgcn_wmma_f32_16x16x4_f32(false, A[ks], false, bm,
                                                    (short)0, acc, false, false);
      }
#pragma unroll
      for (int g = 0; g < 8; ++g) {
        size_t vv = (size_t)(b * NVERT) + vbase + g + row;
        out[vv * OUTW + OFF_F3 + c0 + m] = acc[g];
      }
    }
  }
}

// ---------------- launcher ----------------
extern "C" void kernel_launch(void* const* d_in, const int* in_sizes, int n_in,
                              void* d_out, int out_size, void* d_ws, size_t ws_size,
                              hipStream_t stream) {
  (void)in_sizes; (void)n_in; (void)out_size; (void)ws_size;
  const float* vertices = (const float*)d_in[0];
  const float* f0 = (const float*)d_in[1];
  const float* f1 = (const float*)d_in[2];
  const float* f2 = (const float*)d_in[3];
  const float* f3 = (const float*)d_in[4];
  // d_in[5] = proj_mat: unused by the reference
  float* ws  = (float*)d_ws;
  float* out = (float*)d_out;

  transpose_chw_hwc<<<dim3(512), 256, 0, stream>>>(f0, ws + WS0, 64, 3136);
  transpose_chw_hwc<<<dim3(256), 256, 0, stream>>>(f1, ws + WS1, 128, 784);
  transpose_chw_hwc<<<dim3(128), 256, 0, stream>>>(f2, ws + WS2, 256, 196);
  transpose_chw_hwc<<<dim3(64),  256, 0, stream>>>(f3, ws + WS3, 512, 49);

  dim3 grid(NCHUNK, NBATCH);
  graphproject_kernel<<<grid, 256, 0, stream>>>(vertices, ws, out);
}